// WeightedBCEWithLogitsLoss_65678639890438
// MI455X (gfx1250) — compile-verified
//
#include <hip/hip_runtime.h>

typedef __attribute__((ext_vector_type(2))) float v2f;
typedef __attribute__((ext_vector_type(8))) float v8f;

#define BN 64
#define LN 512
#define CHUNKS 8
#define NW 8
#define SLOTS (CHUNKS * NW)   // 64 wave-slots per sample
#define NT (NW * 32)          // 256 threads = 8 waves (wave32)

__device__ __forceinline__ float wave_reduce_sum(float v) {
#pragma unroll
  for (int m = 16; m >= 1; m >>= 1)
    v += __shfl_xor(v, m, 32);
  return v;
}

// Pass 1: per-(sample, wave-slot) partial sums of targets over the masked region.
__global__ void contacts_kernel(const float* __restrict__ targets,
                                const int* __restrict__ seq_lens,
                                float* __restrict__ ncPart) {
  const int chunk = blockIdx.x % CHUNKS;
  const int b     = blockIdx.x / CHUNKS;
  const int wave  = threadIdx.x >> 5;
  const int lane  = threadIdx.x & 31;
  const int slot  = chunk * NW + wave;
  const int sl    = seq_lens[b];
  const float* tb = targets + (size_t)b * LN * LN;

  float acc = 0.0f;
  for (int r = slot; r < sl; r += SLOTS) {
    const float* row = tb + (size_t)r * LN;
    for (int base = 0; base < sl; base += 128) {     // 32 lanes x float4
      int j4 = base + lane * 4;
      int jc = (j4 > (LN - 4)) ? (LN - 4) : j4;      // clamp: load stays in row
      float4 t = *(const float4*)(row + jc);
      acc += (j4 + 0 < sl) ? t.x : 0.0f;
      acc += (j4 + 1 < sl) ? t.y : 0.0f;
      acc += (j4 + 2 < sl) ? t.z : 0.0f;
      acc += (j4 + 3 < sl) ? t.w : 0.0f;
    }
  }
  acc = wave_reduce_sum(acc);
  if (lane == 0) ncPart[b * SLOTS + slot] = acc;
}

// Pass 2: weighted BCE elem sums, WMMA f32 16x16x4 used as fp32 accumulator.
__global__ void loss_kernel(const float* __restrict__ inputs,
                            const float* __restrict__ targets,
                            const int* __restrict__ seq_lens,
                            const float* __restrict__ ncPart,
                            float* __restrict__ elemPart) {
  const int chunk = blockIdx.x % CHUNKS;
  const int b     = blockIdx.x / CHUNKS;
  const int wave  = threadIdx.x >> 5;
  const int lane  = threadIdx.x & 31;
  const int slot  = chunk * NW + wave;
  const int sl    = seq_lens[b];

  // Reconstruct num_contacts deterministically (fixed order, uniform).
  float nc = 0.0f;
  for (int k = 0; k < SLOTS; ++k) nc += ncPart[b * SLOTS + k];
  const float slf     = (float)sl;
  const float total   = fmaxf(slf * slf, 1.0f);
  const float density = nc / total;
  const float wgt     = fminf(fmaxf(0.01f / (density + 1e-6f), 1.0f), 50.0f);

  const size_t off = (size_t)b * LN * LN;
  const float* xb  = inputs  + off;
  const float* tb  = targets + off;

  const v2f ones = {1.0f, 1.0f};
  v8f acc = {0.0f, 0.0f, 0.0f, 0.0f, 0.0f, 0.0f, 0.0f, 0.0f};

  for (int r = slot; r < sl; r += SLOTS) {
    const float* xrow = xb + (size_t)r * LN;
    const float* trow = tb + (size_t)r * LN;
    for (int base = 0; base < sl; base += 128) {
      int j4 = base + lane * 4;
      int jc = (j4 > (LN - 4)) ? (LN - 4) : j4;
      float4 x4 = *(const float4*)(xrow + jc);
      float4 t4 = *(const float4*)(trow + jc);
      float xs[4] = {x4.x, x4.y, x4.z, x4.w};
      float ts[4] = {t4.x, t4.y, t4.z, t4.w};
      float e[4];
#pragma unroll
      for (int k = 0; k < 4; ++k) {
        float x = xs[k];
        float t = ts[k];
        float a = fabsf(x);
        float l = __logf(1.0f + __expf(-a));      // log1p(exp(-|x|))
        float sp_neg = l + fmaxf(-x, 0.0f);       // softplus(-x)
        float sp_pos = l + fmaxf(x, 0.0f);        // x + softplus(-x) = softplus(x)
        float ek = wgt * t * sp_neg + (1.0f - t) * sp_pos;
        e[k] = (j4 + k < sl) ? ek : 0.0f;         // arithmetic mask, EXEC stays full
      }
      v2f b0 = {e[0], e[1]};
      v2f b1 = {e[2], e[3]};
      // D = ones(16x4) x B(4x16) + C : every row of D accumulates column sums.
      acc = __builtin_amdgcn_wmma_f32_16x16x4_f32(false, ones, false, b0,
                                                  (short)0, acc, false, false);
      acc = __builtin_amdgcn_wmma_f32_16x16x4_f32(false, ones, false, b1,
                                                  (short)0, acc, false, false);
    }
  }
  // acc[0]: lanes 0-15 hold colsums 0..15 (M=0), lanes 16-31 the same (M=8).
  float v = wave_reduce_sum(acc[0]) * 0.5f;
  if (lane == 0) elemPart[b * SLOTS + slot] = v;
}

// Pass 3: deterministic fixed-order final reduction.
__global__ void finalize_kernel(const float* __restrict__ elemPart,
                                const int* __restrict__ seq_lens,
                                float* __restrict__ perS,
                                float* __restrict__ out) {
  int b = threadIdx.x;
  if (b < BN) {
    float s = 0.0f;
    for (int k = 0; k < SLOTS; ++k) s += elemPart[b * SLOTS + k];
    float slf = (float)seq_lens[b];
    perS[b] = s / fmaxf(slf * slf, 1.0f);
  }
  __syncthreads();
  if (threadIdx.x == 0) {
    float s = 0.0f;
    for (int k = 0; k < BN; ++k) s += perS[k];
    out[0] = s * (1.0f / (float)BN);
  }
}

extern "C" void kernel_launch(void* const* d_in, const int* in_sizes, int n_in,
                              void* d_out, int out_size, void* d_ws, size_t ws_size,
                              hipStream_t stream) {
  (void)in_sizes; (void)n_in; (void)out_size; (void)ws_size;
  const float* inputs  = (const float*)d_in[0];
  const float* targets = (const float*)d_in[1];
  const int*   seq     = (const int*)d_in[2];
  float* out = (float*)d_out;
  float* ws  = (float*)d_ws;

  float* ncPart   = ws;                    // BN*SLOTS = 4096 floats
  float* elemPart = ws + BN * SLOTS;       // 4096 floats
  float* perS     = ws + 2 * BN * SLOTS;   // 64 floats

  dim3 grid(BN * CHUNKS), block(NT);
  contacts_kernel<<<grid, block, 0, stream>>>(targets, seq, ncPart);
  loss_kernel<<<grid, block, 0, stream>>>(inputs, targets, seq, ncPart, elemPart);
  finalize_kernel<<<1, 64, 0, stream>>>(elemPart, seq, perS, out);
}